// DirectedHyperConvNetwork_20358144983741
// MI455X (gfx1250) — compile-verified
//
#include <hip/hip_runtime.h>
#include <hip/hip_bf16.h>
#include <stdint.h>

// ---------------------------------------------------------------------------
// DirectedHyperConvNetwork for MI455X (gfx1250).
//
// Per layer:  tmpH = SpMM(tar, x)        [H,256]   (atomic-scatter COO SpMM)
//             tmpN = SpMM(src, tmpH)     [N,256]
//             x    = dropout(tmpN + x)   (inverted dropout, keep=0.9)
// Output:     mean(poi_embs, x1, x2, x3) -- computed with V_WMMA_F32_16X16X4_F32
//             (A = 16x4 matrix of 0.25, B = the 4 finals -> D rows all hold the
//              exact f32 average; 64 floats read : 16 written per WMMA issue).
//
// All gathers/scatters are L2-resident (X = 51.2 MB << 192 MB L2), so the SpMM
// uses wave-per-nonzero float4 gathers + non-returning global_atomic_add_f32.
// ---------------------------------------------------------------------------

#define D 256
#define KEEP_THRESH 3865470566u   /* floor(0.9 * 2^32) */
#define INV_KEEP (1.0f / 0.9f)

typedef __attribute__((ext_vector_type(2))) float v2f;
typedef __attribute__((ext_vector_type(8))) float v8f;

__device__ __forceinline__ uint32_t hash_u32(uint32_t x) {
  x ^= x >> 16; x *= 0x7feb352du;
  x ^= x >> 15; x *= 0x846ca68bu;
  x ^= x >> 16;
  return x;
}

// ---- zero a float buffer (float4 stores, grid-stride) ----------------------
__global__ void zero_f32(float4* __restrict__ p, int n4) {
  int i = blockIdx.x * blockDim.x + threadIdx.x;
  int stride = gridDim.x * blockDim.x;
  float4 z; z.x = 0.f; z.y = 0.f; z.z = 0.f; z.w = 0.f;
  for (; i < n4; i += stride) p[i] = z;
}

// ---- COO SpMM: out[rows[e]] += vals[e] * X[cols[e]]  (wave per nonzero) ----
__global__ void spmm_coo_atomic(const int* __restrict__ rows,
                                const int* __restrict__ cols,
                                const float* __restrict__ vals,
                                const float* __restrict__ X,
                                float* __restrict__ out, int nnz) {
  const int lane  = threadIdx.x & 31;
  int e           = (blockIdx.x * blockDim.x + threadIdx.x) >> 5;
  const int nwav  = (gridDim.x * blockDim.x) >> 5;
  for (; e < nnz; e += nwav) {
    // uniform per wave (broadcast through scalar regs)
    const int   r = __builtin_amdgcn_readfirstlane(rows[e]);
    const int   c = __builtin_amdgcn_readfirstlane(cols[e]);
    const float v = vals[e];
    const float* __restrict__ xr = X   + (size_t)c * D;
    float*       __restrict__ orow = out + (size_t)r * D;
#pragma unroll
    for (int j = 0; j < 2; ++j) {
      const int off = j * 128 + lane * 4;          // 32 lanes x float4 x 2
      float4 xv = *(const float4*)(xr + off);      // global_load_b128 (L2 hit)
      float* op = orow + off;
      // non-returning f32 atomic adds resolved at L2
      __hip_atomic_fetch_add(op + 0, v * xv.x, __ATOMIC_RELAXED, __HIP_MEMORY_SCOPE_AGENT);
      __hip_atomic_fetch_add(op + 1, v * xv.y, __ATOMIC_RELAXED, __HIP_MEMORY_SCOPE_AGENT);
      __hip_atomic_fetch_add(op + 2, v * xv.z, __ATOMIC_RELAXED, __HIP_MEMORY_SCOPE_AGENT);
      __hip_atomic_fetch_add(op + 3, v * xv.w, __ATOMIC_RELAXED, __HIP_MEMORY_SCOPE_AGENT);
    }
  }
}

// ---- residual + inverted dropout:  fnew = mask ? (msg+fprev)/keep : 0 ------
__global__ void residual_dropout(const float4* __restrict__ msg,
                                 const float4* __restrict__ fprev,
                                 float4* __restrict__ fnew,
                                 int n4, uint32_t seed) {
  int i = blockIdx.x * blockDim.x + threadIdx.x;
  int stride = gridDim.x * blockDim.x;
  for (; i < n4; i += stride) {
    float4 m = msg[i];
    float4 p = fprev[i];
    uint32_t base = (uint32_t)i * 4u;
    float4 o;
    o.x = (hash_u32((base + 0u) ^ seed) < KEEP_THRESH) ? (m.x + p.x) * INV_KEEP : 0.f;
    o.y = (hash_u32((base + 1u) ^ seed) < KEEP_THRESH) ? (m.y + p.y) * INV_KEEP : 0.f;
    o.z = (hash_u32((base + 2u) ^ seed) < KEEP_THRESH) ? (m.z + p.z) * INV_KEEP : 0.f;
    o.w = (hash_u32((base + 3u) ^ seed) < KEEP_THRESH) ? (m.w + p.w) * INV_KEEP : 0.f;
    fnew[i] = o;
  }
}

// ---- mean of 4 tensors via V_WMMA_F32_16X16X4_F32 --------------------------
// B (4x16 f32, 2 VGPRs): VGPR0 lanes0-15 = K0 (f0), lanes16-31 = K2 (f2);
//                        VGPR1 lanes0-15 = K1 (f1), lanes16-31 = K3 (f3).
// A = 0.25 everywhere -> every row of D = exact f32 average of the 4 finals.
__global__ void mean4_wmma(const float* __restrict__ f0,
                           const float* __restrict__ f1,
                           const float* __restrict__ f2,
                           const float* __restrict__ f3,
                           float* __restrict__ out, int nchunks /* total/16 */) {
  const int lane  = threadIdx.x & 31;
  int chunk       = (blockIdx.x * blockDim.x + threadIdx.x) >> 5;
  const int nwav  = (gridDim.x * blockDim.x) >> 5;
  const int n     = lane & 15;
  const float* srcLo = (lane < 16) ? f0 : f2;   // K=0 / K=2
  const float* srcHi = (lane < 16) ? f1 : f3;   // K=1 / K=3
  for (; chunk < nchunks; chunk += nwav) {
    const int base = chunk * 16;
    v2f b; b.x = srcLo[base + n]; b.y = srcHi[base + n];
    v2f a; a.x = 0.25f; a.y = 0.25f;
    v8f c = {};
    v8f d = __builtin_amdgcn_wmma_f32_16x16x4_f32(
        /*neg_a=*/false, a, /*neg_b=*/false, b,
        /*c_mod=*/(short)0, c, /*reuse_a=*/false, /*reuse_b=*/false);
    if (lane < 16) out[base + n] = d[0];        // row M=0 holds the average
  }
}

extern "C" void kernel_launch(void* const* d_in, const int* in_sizes, int n_in,
                              void* d_out, int out_size, void* d_ws, size_t ws_size,
                              hipStream_t stream) {
  const float* poi     = (const float*)d_in[0];
  const int*   src_row = (const int*)d_in[1];
  const int*   src_col = (const int*)d_in[2];
  const float* src_val = (const float*)d_in[3];
  const int*   tar_row = (const int*)d_in[4];
  const int*   tar_col = (const int*)d_in[5];
  const float* tar_val = (const float*)d_in[6];

  const int nnz = in_sizes[1];
  const int ND  = in_sizes[0];        // N * 256  (H == N in this problem)
  const int n4  = ND / 4;

  float* tmpH = (float*)d_ws;         // [H,256] hyperedge messages
  float* tmpN = tmpH + (size_t)ND;    // [N,256] node messages
  float* f1   = tmpN + (size_t)ND;    // finals[1]
  float* f2   = f1   + (size_t)ND;    // finals[2]
  float* f3   = (float*)d_out;        // finals[3] lives in d_out, averaged in place

  const float* fin_r[4] = {poi, f1, f2, f3};
  float*       fin_w[4] = {nullptr, f1, f2, f3};

  const int ZB = (n4 + 255) / 256;                 // zero / elementwise blocks
  const int SB = (nnz + 7) / 8;                    // 8 waves/block, wave per nnz

  for (int layer = 0; layer < 3; ++layer) {
    zero_f32<<<ZB, 256, 0, stream>>>((float4*)tmpH, n4);
    spmm_coo_atomic<<<SB, 256, 0, stream>>>(tar_row, tar_col, tar_val,
                                            fin_r[layer], tmpH, nnz);
    zero_f32<<<ZB, 256, 0, stream>>>((float4*)tmpN, n4);
    spmm_coo_atomic<<<SB, 256, 0, stream>>>(src_row, src_col, src_val,
                                            tmpH, tmpN, nnz);
    residual_dropout<<<ZB, 256, 0, stream>>>((const float4*)tmpN,
                                             (const float4*)fin_r[layer],
                                             (float4*)fin_w[layer + 1], n4,
                                             0x9E3779B9u * (uint32_t)(layer + 1));
  }

  const int nchunks = ND / 16;
  const int AB = (nchunks * 32 + 255) / 256;       // one wave per 16-elem chunk
  mean4_wmma<<<AB, 256, 0, stream>>>(poi, f1, f2, f3, (float*)d_out, nchunks);
}